// ViterbiDecoder_70531952935328
// MI455X (gfx1250) — compile-verified
//
#include <hip/hip_runtime.h>
#include <hip/hip_bf16.h>
#include <stdint.h>

// Softmax-smoothed Viterbi forward, anti-diagonal wavefront, tiled 8x8.
// N=M=1024, S=4. Tile T=128. 15 stream-ordered launches (one per tile-diagonal).

#define NEGV (-1.0e8f)

constexpr int NN = 1024;
constexpr int MM = 1024;
constexpr int T  = 128;          // tile edge
constexpr int K  = NN / T;       // 8 tiles per side

// Inter-tile boundary storage (written/read within one kernel_launch sequence,
// strictly producer-before-consumer via stream-ordered launches).
__device__ float4 g_Brow[K][K][T];  // bottom row V of tile (ti,tj)
__device__ float4 g_Bcol[K][K][T];  // right col V of tile (ti,tj)
__device__ float4 g_Bcor[K][K];     // bottom-right corner V of tile (ti,tj)

__device__ __forceinline__ float4 mk4(float v) {
  float4 r; r.x = v; r.y = v; r.z = v; r.w = v; return r;
}

__device__ __forceinline__ float lse4(float a, float b, float c, float d) {
  float m = fmaxf(fmaxf(a, b), fmaxf(c, d));
  return m + logf(expf(a - m) + expf(b - m) + expf(c - m) + expf(d - m));
}

// Async global->LDS 16-byte copy (CDNA5 ASYNCcnt path), per-lane addresses.
__device__ __forceinline__ void async_ld_b128(unsigned lds_off, const void* gptr) {
  unsigned long long ga = (unsigned long long)(uintptr_t)gptr;
  asm volatile("global_load_async_to_lds_b128 %0, %1, off"
               :: "v"(lds_off), "v"(ga) : "memory");
}
__device__ __forceinline__ void wait_async0() {
  asm volatile("s_wait_asynccnt 0" ::: "memory");
}

__global__ __launch_bounds__(T)
void viterbi_tile_kernel(const float* __restrict__ theta,
                         const float* __restrict__ A,
                         float* __restrict__ out,
                         int td) {
  __shared__ float4 Vb[3][T + 2];   // rotating diagonal buffers; entry idx = li+1
  __shared__ float4 thS[2][T];      // double-buffered theta (async-filled)
  __shared__ float4 topRow[T];      // V[ti*T][tj*T+1+k]
  __shared__ float4 leftCol[T];     // V[ti*T+1+k][tj*T]

  const int li    = threadIdx.x;
  const int tiMin = (td - (K - 1) > 0) ? td - (K - 1) : 0;
  const int ti    = tiMin + (int)blockIdx.x;
  const int tj    = td - ti;

  // ---- load tile boundary inputs ----
  topRow[li]  = (ti == 0) ? mk4(NEGV) : g_Brow[ti - 1][tj][li];
  leftCol[li] = (tj == 0) ? mk4(NEGV) : g_Bcol[ti][tj - 1][li];

  float4 cornerV;
  if (ti == 0 && tj == 0)      cornerV = mk4(0.0f);
  else if (ti == 0 || tj == 0) cornerV = mk4(NEGV);
  else                         cornerV = g_Bcor[ti - 1][tj - 1];

  if (li == 0) {
    // before ld=0: prev buffer = Vb[2], prev2 buffer = Vb[1]
    Vb[2][0] = topRow[0];    // V(-1, 0)
    Vb[2][1] = leftCol[0];   // V(0, -1)
    Vb[1][0] = cornerV;      // V(-1, -1)
  }

  const int r  = ti * T + li;        // global row index (i-1), always in [0,N)
  const int c0 = tj * T;

  // clamped flat cell index (r*M + c) for inner diagonal ld_
  auto cellIdx = [&](int ld_) -> size_t {
    int c = c0 + (ld_ - li);
    c = c < 0 ? 0 : (c > MM - 1 ? MM - 1 : c);
    return (size_t)r * MM + (size_t)c;
  };

  // ---- prefetch for ld = 0 ----
  float4 A0, A1, A2, A3;
  {
    size_t idx = cellIdx(0);
    const float4* ap = reinterpret_cast<const float4*>(A + idx * 16);
    A0 = ap[0]; A1 = ap[1]; A2 = ap[2]; A3 = ap[3];
    unsigned lo = (unsigned)(uintptr_t)(&thS[0][li]);
    async_ld_b128(lo, theta + idx * 4);
  }
  __syncthreads();

  // ---- wavefront over 2T-1 inner diagonals ----
  for (int ld = 0; ld < 2 * T - 1; ++ld) {
    const int cur   = ld % 3;
    const int prev  = (ld + 2) % 3;
    const int prev2 = (ld + 1) % 3;
    const int lj    = ld - li;
    const bool active = (lj >= 0) && (lj < T);

    // theta for current diagonal has landed in LDS
    wait_async0();

    // issue async theta for next diagonal (clamped address; all lanes issue)
    {
      size_t idx = cellIdx(ld + 1);
      unsigned lo = (unsigned)(uintptr_t)(&thS[(ld + 1) & 1][li]);
      async_ld_b128(lo, theta + idx * 4);
    }

    // prefetch A for next diagonal into registers
    float4 B0, B1, B2, B3;
    {
      size_t idx = cellIdx(ld + 1);
      const float4* ap = reinterpret_cast<const float4*>(A + idx * 16);
      B0 = ap[0]; B1 = ap[1]; B2 = ap[2]; B3 = ap[3];
    }

    if (active) {
      float4 vd = Vb[prev2][li];       // V(li-1, lj-1)
      float4 vu = Vb[prev][li];        // V(li-1, lj)
      float4 vl = Vb[prev][li + 1];    // V(li,   lj-1)
      float4 th = thS[ld & 1][li];

      float4 o;
      o.x = lse4(vd.x + A0.x, vd.y + A0.y, vd.z + A0.z, vd.w + A0.w) + th.x; // m
      o.y = lse4(vu.x + A1.x, vu.y + A1.y, vu.z + A1.z, vu.w + A1.w) + th.y; // x
      o.z = lse4(vl.x + A2.x, vl.y + A2.y, vl.z + A2.z, vl.w + A2.w) + th.z; // y
      o.w = lse4(vd.x + A3.x, vd.y + A3.y, vd.z + A3.z, vd.w + A3.w) + th.w; // s

      Vb[cur][li + 1] = o;

      if (li == T - 1) g_Brow[ti][tj][lj] = o;
      if (lj == T - 1) g_Bcol[ti][tj][li] = o;
      if (li == T - 1 && lj == T - 1) {
        g_Bcor[ti][tj] = o;
        if (ti == K - 1 && tj == K - 1)
          out[0] = lse4(o.x, o.y, o.z, o.w);   // lse(V[N,M])
      }
    }

    // boundary presets for the buffer that becomes "prev" next iteration
    if (li == 0)
      Vb[cur][0] = (ld + 1 < T) ? topRow[ld + 1] : mk4(NEGV);
    if (li == ld + 1 && li < T)
      Vb[cur][li + 1] = leftCol[li];

    A0 = B0; A1 = B1; A2 = B2; A3 = B3;
    __syncthreads();
  }
}

extern "C" void kernel_launch(void* const* d_in, const int* in_sizes, int n_in,
                              void* d_out, int out_size, void* d_ws, size_t ws_size,
                              hipStream_t stream) {
  (void)in_sizes; (void)n_in; (void)out_size; (void)d_ws; (void)ws_size;
  const float* theta = (const float*)d_in[0];  // [1024,1024,4]
  const float* A     = (const float*)d_in[1];  // [1024,1024,4,4]
  float* out         = (float*)d_out;          // scalar

  for (int td = 0; td < 2 * K - 1; ++td) {
    int tiMin = (td - (K - 1) > 0) ? td - (K - 1) : 0;
    int tiMax = (td < K - 1) ? td : K - 1;
    int cnt   = tiMax - tiMin + 1;
    viterbi_tile_kernel<<<cnt, T, 0, stream>>>(theta, A, out, td);
  }
}